// NMS_62869731279593
// MI455X (gfx1250) — compile-verified
//
#include <hip/hip_runtime.h>

#define R_CHK   144
#define N_VAR   576
#define BATCH   512
#define ITERS   3
#define NLEV    16
#define ROW_DEG 15

#define WAVES_PER_BLOCK 8
#define BLOCK_THREADS   (WAVES_PER_BLOCK * 32)
#define ROW_SLOTS       ((R_CHK + 31) / 32)   // 5 row slots per lane (wave32)

typedef __attribute__((address_space(3))) float lds_float;

// gfx1250 async global->LDS copy (ASYNCcnt-tracked, bypasses VGPRs)
__device__ __forceinline__ void async_copy_b32(lds_float* ldst, const float* gsrc) {
  asm volatile("global_load_async_to_lds_b32 %0, %1, off"
               :: "v"(ldst), "v"(gsrc)
               : "memory");
}

__device__ __forceinline__ void wait_async_zero() {
  asm volatile("s_wait_asynccnt 0x0" ::: "memory");
}

__device__ __forceinline__ float signf(float x) {
  return (x > 0.0f) ? 1.0f : ((x < 0.0f) ? -1.0f : 0.0f);
}

// qk[searchsorted(bnd, y, side='left')] == qk[#\{bnd[i] < y\}]
__device__ __forceinline__ float quantf(float y, const float* q, const float* bnd) {
  float v = q[0];
  #pragma unroll
  for (int i = 0; i < NLEV - 1; ++i)
    v = (y > bnd[i]) ? q[i + 1] : v;
  return v;
}

__global__ __launch_bounds__(BLOCK_THREADS, 1)
void ldpc_minsum_kernel(const float* __restrict__ r,
                        const float* __restrict__ alpha,
                        const float* __restrict__ beta,
                        const float* __restrict__ eta,
                        const float* __restrict__ qk,
                        const float* __restrict__ H,
                        float* __restrict__ out)
{
  __shared__ int   s_cols[R_CHK * ROW_DEG];          //  8.6 KB  sparse structure of H
  __shared__ float s_r  [WAVES_PER_BLOCK * N_VAR];   // 18.4 KB  per-wave received vector
  __shared__ float s_col[WAVES_PER_BLOCK * N_VAR];   // 18.4 KB  per-wave column sums

  const int tid  = threadIdx.x;
  const int wave = tid >> 5;
  const int lane = tid & 31;
  const int b    = blockIdx.x * WAVES_PER_BLOCK + wave;   // 64 blocks * 8 waves == BATCH

  float* srw = &s_r[wave * N_VAR];
  float* cw  = &s_col[wave * N_VAR];

  // 1) async-stage r[b, :] into this wave's LDS slab (18 dwords per lane, all lanes active)
  {
    const float* rb = r + b * N_VAR;
    #pragma unroll
    for (int i = 0; i < N_VAR / 32; ++i) {
      const int v = i * 32 + lane;
      async_copy_b32((lds_float*)&srw[v], rb + v);
    }
  }

  // 2) build per-row column-index table from dense H (once per block, shared)
  if (tid < R_CHK) {
    const float* hrow = H + tid * N_VAR;
    int n = 0;
    for (int v = 0; v < N_VAR; ++v) {
      if (hrow[v] > 0.5f) {
        if (n < ROW_DEG) s_cols[tid * ROW_DEG + n] = v;
        ++n;
      }
    }
  }

  // quantizer levels + decision boundaries (uniform loads -> scalarized)
  float q[NLEV], bnd[NLEV - 1];
  #pragma unroll
  for (int i = 0; i < NLEV; ++i) q[i] = qk[i];
  #pragma unroll
  for (int i = 0; i < NLEV - 1; ++i) bnd[i] = 0.5f * (q[i] + q[i + 1]);

  __syncthreads();     // s_cols visible block-wide
  wait_async_zero();   // this wave's s_r slab resident in LDS

  // 3) init messages M[row][k] = r[b, col(row,k)]; rows striped lane + 32*s
  float M[ROW_SLOTS][ROW_DEG];
  #pragma unroll
  for (int s = 0; s < ROW_SLOTS; ++s) {
    const int row = lane + 32 * s;
    if (row < R_CHK) {
      #pragma unroll
      for (int k = 0; k < ROW_DEG; ++k)
        M[s][k] = srw[s_cols[row * ROW_DEG + k]];
    }
  }

  float m1s[ROW_SLOTS], m2s[ROW_SLOTS], rss[ROW_SLOTS];

  #pragma unroll
  for (int it = 0; it < ITERS; ++it) {
    const float al = alpha[it];
    const float be = beta[it];
    const float et = eta[it];

    // zero this wave's column accumulator
    #pragma unroll
    for (int i = 0; i < N_VAR / 32; ++i) cw[i * 32 + lane] = 0.0f;
    __syncthreads();

    // check-node update: two-min + sign product, quantize E, scatter-add columns
    #pragma unroll
    for (int s = 0; s < ROW_SLOTS; ++s) {
      const int row = lane + 32 * s;
      if (row < R_CHK) {
        float m1 = __builtin_inff(), m2 = __builtin_inff(), rs = 1.0f;
        #pragma unroll
        for (int k = 0; k < ROW_DEG; ++k) {
          const float m = M[s][k];
          const float a = fabsf(m);
          rs *= signf(m);
          if (a < m1)      { m2 = m1; m1 = a; }
          else if (a < m2) { m2 = a; }
        }
        m1s[s] = m1; m2s[s] = m2; rss[s] = rs;
        #pragma unroll
        for (int k = 0; k < ROW_DEG; ++k) {
          const float m   = M[s][k];
          const float a   = fabsf(m);
          const float mex = (a == m1) ? m2 : m1;      // min-excluding-self (tie-exact)
          const float E   = al * rs * signf(m) * fmaxf(mex - be, 0.0f);
          const float Eq  = quantf(et * E, q, bnd);
          atomicAdd(&cw[s_cols[row * ROW_DEG + k]], Eq);   // ds_add_f32
        }
      }
    }
    __syncthreads();

    // variable-node update: M = quant(eta * (r + col - E)), E recomputed from (m1,m2,rs)
    #pragma unroll
    for (int s = 0; s < ROW_SLOTS; ++s) {
      const int row = lane + 32 * s;
      if (row < R_CHK) {
        const float m1 = m1s[s], m2 = m2s[s], rs = rss[s];
        #pragma unroll
        for (int k = 0; k < ROW_DEG; ++k) {
          const int   v   = s_cols[row * ROW_DEG + k];
          const float m   = M[s][k];
          const float a   = fabsf(m);
          const float mex = (a == m1) ? m2 : m1;
          const float E   = al * rs * signf(m) * fmaxf(mex - be, 0.0f);
          const float Eq  = quantf(et * E, q, bnd);
          M[s][k] = quantf(et * (srw[v] + cw[v] - Eq), q, bnd);
        }
      }
    }
    __syncthreads();
  }

  // 4) out[b,:] = r[b,:] + colsum(E_last)  (coalesced 128B stores per wave)
  #pragma unroll
  for (int i = 0; i < N_VAR / 32; ++i) {
    const int v = i * 32 + lane;
    out[b * N_VAR + v] = srw[v] + cw[v];
  }
}

extern "C" void kernel_launch(void* const* d_in, const int* in_sizes, int n_in,
                              void* d_out, int out_size, void* d_ws, size_t ws_size,
                              hipStream_t stream) {
  (void)in_sizes; (void)n_in; (void)d_ws; (void)ws_size; (void)out_size;
  // setup_inputs() order: r, alpha, beta, eta, qk, H
  const float* r     = (const float*)d_in[0];
  const float* alpha = (const float*)d_in[1];
  const float* beta  = (const float*)d_in[2];
  const float* eta   = (const float*)d_in[3];
  const float* qk    = (const float*)d_in[4];
  const float* H     = (const float*)d_in[5];
  float* out = (float*)d_out;

  dim3 grid(BATCH / WAVES_PER_BLOCK);   // 64 blocks, 8 wave32s each -> 512 codewords
  ldpc_minsum_kernel<<<grid, BLOCK_THREADS, 0, stream>>>(r, alpha, beta, eta, qk, H, out);
}